// TGNNPO_26113401159680
// MI455X (gfx1250) — compile-verified
//
#include <hip/hip_runtime.h>
#include <math.h>

#define NB 20000
#define BB 4
#define FIN 8
#define PP 12
#define OUTF 32

typedef __attribute__((ext_vector_type(2))) float v2f;
typedef __attribute__((ext_vector_type(8))) float v8f;

// ---------------- init: zero hacc + per-node edge counters ----------------
__global__ void tg_init(float* __restrict__ hacc, int* __restrict__ count, int Nn) {
  int idx = blockIdx.x * blockDim.x + threadIdx.x;
  if (idx < Nn * 128) hacc[idx] = 0.f;
  if (idx < Nn) count[idx] = 0;
}

// ---------------- CSR build: count in-degree ----------------
__global__ void tg_count(const int* __restrict__ ei, int* __restrict__ count, int Ee) {
  int e = blockIdx.x * blockDim.x + threadIdx.x;
  if (e < Ee) atomicAdd(&count[ei[Ee + e]], 1);
}

// ---------------- CSR build: exclusive scan + dinv ----------------
__global__ void tg_scan(const int* __restrict__ count, int* __restrict__ row_start,
                        int* __restrict__ cursor, float* __restrict__ dinv,
                        int Nn, int Ee) {
  __shared__ int part[256];
  __shared__ int offs[256];
  int t = threadIdx.x;
  int chunk = (Nn + 255) / 256;
  int lo = t * chunk, hi = lo + chunk; if (hi > Nn) hi = Nn;
  int s = 0;
  for (int i = lo; i < hi; ++i) s += count[i];
  part[t] = s;
  __syncthreads();
  if (t == 0) {
    int run = 0;
    for (int i = 0; i < 256; ++i) { offs[i] = run; run += part[i]; }
  }
  __syncthreads();
  int base = offs[t];
  for (int i = lo; i < hi; ++i) {
    row_start[i] = base;
    cursor[i] = base;
    int c = count[i];
    dinv[i] = rsqrtf((float)(c + 1));   // self-loop included in degree
    base += c;
  }
  if (t == 0) row_start[Nn] = Ee;
}

// ---------------- CSR build: fill column (src) indices ----------------
__global__ void tg_fill(const int* __restrict__ ei, int* __restrict__ cursor,
                        int* __restrict__ col, int Ee) {
  int e = blockIdx.x * blockDim.x + threadIdx.x;
  if (e < Ee) {
    int s = ei[e], d = ei[Ee + e];
    int pos = atomicAdd(&cursor[d], 1);
    col[pos] = s;
  }
}

// ---------------- prep: fuse conv*lin weights, biases, softmax(attention) ----------------
__global__ void tg_prep(const float* __restrict__ czw, const float* __restrict__ czb,
                        const float* __restrict__ lzw, const float* __restrict__ lzb,
                        const float* __restrict__ chw, const float* __restrict__ chb,
                        const float* __restrict__ lhw, const float* __restrict__ lhb,
                        const float* __restrict__ attention,
                        float* __restrict__ Wcomb, float* __restrict__ biasC,
                        float* __restrict__ probs) {
  int t = threadIdx.x;        // 0..511
  int k = t >> 6;             // 0..7  (F_IN)
  int f = t & 63;             // 0..63 (z:0-31, h:32-63)
  int g = f >> 5;
  int o = f & 31;
  const float* cw = g ? chw : czw;   // (8,32)
  const float* lw = g ? lhw : lzw;   // (64,32), rows 0..31 used (H0==0 kills rows 32..63)
  float acc = 0.f;
  for (int j = 0; j < 32; ++j) acc += cw[k * 32 + j] * lw[j * 32 + o];
  Wcomb[k * 64 + f] = acc;
  if (k == 0) {
    const float* cb = g ? chb : czb;
    const float* lb = g ? lhb : lzb;
    float bacc = lb[o];
    for (int j = 0; j < 32; ++j) bacc += cb[j] * lw[j * 32 + o];
    biasC[f] = bacc;
  }
  if (t == 0) {
    float m = attention[0];
    for (int p = 1; p < PP; ++p) m = fmaxf(m, attention[p]);
    float e[PP]; float sum = 0.f;
    for (int p = 0; p < PP; ++p) { e[p] = __expf(attention[p] - m); sum += e[p]; }
    float inv = 1.f / sum;
    for (int p = 0; p < PP; ++p) probs[p] = e[p] * inv;
  }
}

// ---------------- U = Xp @ Wcomb via V_WMMA_F32_16X16X4_F32 ----------------
// rows = (n*B + b) pairs (80000), K = 8, cols = 64. One wave per 16-row tile,
// 4 column tiles x 2 K-steps = 8 chained WMMAs. EXEC all-ones (no divergence).
__global__ void __launch_bounds__(256) tg_u(const float* __restrict__ x,
                                            const float* __restrict__ Wcomb,
                                            float* __restrict__ U,
                                            int Nn, int period) {
  __shared__ float Wl[512];
  int tid = threadIdx.x;
  for (int i = tid; i < 512; i += 256) Wl[i] = Wcomb[i];
  __syncthreads();
  int wave = tid >> 5;
  int lane = tid & 31;
  int rowBase = (blockIdx.x * 8 + wave) * 16;
  int r   = lane & 15;
  int kh  = lane >> 4;                 // which K-pair this lane holds
  int row = rowBase + r;               // row = n*B + b
  int n = row >> 2, b = row & 3;
  const float* xp = x + ((size_t)(b * Nn + n) * FIN) * PP + period;

#if __has_builtin(__builtin_amdgcn_wmma_f32_16x16x4_f32)
  v8f c0 = {0.f,0.f,0.f,0.f,0.f,0.f,0.f,0.f};
  v8f c1 = c0, c2 = c0, c3 = c0;
  #pragma unroll
  for (int ks = 0; ks < 2; ++ks) {
    int k0 = ks * 4 + 2 * kh;
    v2f a;
    a.x = xp[(size_t)k0 * PP];                 // A: lane=M, VGPR0/1 = K pair
    a.y = xp[(size_t)(k0 + 1) * PP];
    const float* wb = &Wl[k0 * 64];
    int cN = lane & 15;
    v2f b0, b1, b2, b3;
    b0.x = wb[cN];      b0.y = wb[64 + cN];    // B: lane=N, VGPR0/1 = K pair
    b1.x = wb[16 + cN]; b1.y = wb[80 + cN];
    b2.x = wb[32 + cN]; b2.y = wb[96 + cN];
    b3.x = wb[48 + cN]; b3.y = wb[112 + cN];
    c0 = __builtin_amdgcn_wmma_f32_16x16x4_f32(false, a, false, b0, (short)0, c0, false, false);
    c1 = __builtin_amdgcn_wmma_f32_16x16x4_f32(false, a, false, b1, (short)0, c1, false, false);
    c2 = __builtin_amdgcn_wmma_f32_16x16x4_f32(false, a, false, b2, (short)0, c2, false, false);
    c3 = __builtin_amdgcn_wmma_f32_16x16x4_f32(false, a, false, b3, (short)0, c3, false, false);
  }
  int cN = lane & 15;
  int rOff = 8 * kh;
  #pragma unroll
  for (int j = 0; j < 8; ++j) {                // C: VGPR j -> row j (lanes<16) / j+8
    float* up = U + (size_t)(rowBase + rOff + j) * 64;
    up[cN]      = c0[j];
    up[16 + cN] = c1[j];
    up[32 + cN] = c2[j];
    up[48 + cN] = c3[j];
  }
#else
  // Scalar fallback (keeps correctness if builtin absent; wmma=0 signals retry)
  float xv[FIN];
  #pragma unroll
  for (int k = 0; k < FIN; ++k) xv[k] = xp[(size_t)k * PP];
  for (int f = 0; f < 64; ++f) {
    float acc = 0.f;
    #pragma unroll
    for (int k = 0; k < FIN; ++k) acc += xv[k] * Wl[k * 64 + f];
    U[(size_t)row * 64 + f] = acc;
  }
  (void)kh; (void)r;
#endif
}

// ---------------- fused Â-propagation + gate + attention accumulate ----------------
// One block per dst node; 128 threads = (b,o). CSR gather, no float atomics.
__global__ void __launch_bounds__(128) tg_prop(const float* __restrict__ U,
                                               const int* __restrict__ row_start,
                                               const int* __restrict__ col,
                                               const float* __restrict__ dinv,
                                               const float* __restrict__ biasC,
                                               const float* __restrict__ probs,
                                               float* __restrict__ hacc, int period) {
  int d = blockIdx.x;
  int t = threadIdx.x;            // 0..127
  int b = t >> 5, o = t & 31;
  int fz = b * 64 + o;
  float dd = dinv[d];
  const float* Ud = U + (size_t)d * 256;
  float az = dd * Ud[fz];         // self-loop (dinv^2 applied at the end)
  float ah = dd * Ud[fz + 32];
  int e0 = row_start[d], e1 = row_start[d + 1];
  for (int e = e0; e < e1; ++e) {
    int s = col[e];
    float ws = dinv[s];
    const float* Us = U + (size_t)s * 256;
    az += ws * Us[fz];
    ah += ws * Us[fz + 32];
  }
  float yz = dd * az + biasC[o];
  float yh = dd * ah + biasC[32 + o];
  float Z  = 1.f / (1.f + __expf(-yz));
  float Ht = tanhf(yh);
  hacc[(size_t)d * 128 + t] += probs[period] * (1.f - Z) * Ht;
}

// ---------------- final: relu, out projection, softmax over batch ----------------
__global__ void tg_final(const float* __restrict__ hacc, const float* __restrict__ out_w,
                         const float* __restrict__ out_b, float* __restrict__ out, int Nn) {
  int idx = blockIdx.x * blockDim.x + threadIdx.x;
  if (idx >= Nn * PP) return;
  int n = idx / PP, k = idx % PP;
  float bk = out_b[k];
  float lg[BB];
  #pragma unroll
  for (int b = 0; b < BB; ++b) {
    float acc = bk;
    const float* hp = hacc + (size_t)n * 128 + b * 32;
    #pragma unroll
    for (int o = 0; o < OUTF; ++o) acc += fmaxf(hp[o], 0.f) * out_w[o * PP + k];
    lg[b] = acc;
  }
  float m = fmaxf(fmaxf(lg[0], lg[1]), fmaxf(lg[2], lg[3]));
  float e0 = __expf(lg[0] - m), e1 = __expf(lg[1] - m);
  float e2 = __expf(lg[2] - m), e3 = __expf(lg[3] - m);
  float inv = 1.f / (e0 + e1 + e2 + e3);
  out[((size_t)0 * Nn + n) * PP + k] = e0 * inv;
  out[((size_t)1 * Nn + n) * PP + k] = e1 * inv;
  out[((size_t)2 * Nn + n) * PP + k] = e2 * inv;
  out[((size_t)3 * Nn + n) * PP + k] = e3 * inv;
}

extern "C" void kernel_launch(void* const* d_in, const int* in_sizes, int n_in,
                              void* d_out, int out_size, void* d_ws, size_t ws_size,
                              hipStream_t stream) {
  const float* x    = (const float*)d_in[0];
  const int*   ei   = (const int*)d_in[1];
  const float* czw  = (const float*)d_in[2];
  const float* czb  = (const float*)d_in[3];
  const float* lzw  = (const float*)d_in[4];
  const float* lzb  = (const float*)d_in[5];
  // d_in[6..9] = r-gate: provably unused (H0 == 0)
  const float* chw  = (const float*)d_in[10];
  const float* chb  = (const float*)d_in[11];
  const float* lhw  = (const float*)d_in[12];
  const float* lhb  = (const float*)d_in[13];
  const float* attn = (const float*)d_in[14];
  const float* outw = (const float*)d_in[15];
  const float* outb = (const float*)d_in[16];
  float* out = (float*)d_out;

  const int Nn = NB;
  const int Ee = in_sizes[1] / 2;

  // workspace bump allocator (256B aligned)
  char* ws = (char*)d_ws;
  size_t off = 0;
  auto alloc = [&](size_t bytes) -> void* {
    off = (off + 255) & ~(size_t)255;
    void* p = ws + off;
    off += bytes;
    return p;
  };
  float* hacc      = (float*)alloc((size_t)Nn * 128 * 4);
  float* U         = (float*)alloc((size_t)Nn * 256 * 4);
  int*   count     = (int*)  alloc((size_t)Nn * 4);
  int*   row_start = (int*)  alloc((size_t)(Nn + 1) * 4);
  int*   cursor    = (int*)  alloc((size_t)Nn * 4);
  float* dinv      = (float*)alloc((size_t)Nn * 4);
  int*   col       = (int*)  alloc((size_t)Ee * 4);
  float* Wcomb     = (float*)alloc(512 * 4);
  float* biasC     = (float*)alloc(64 * 4);
  float* probs     = (float*)alloc(PP * 4);

  tg_init<<<(Nn * 128 + 255) / 256, 256, 0, stream>>>(hacc, count, Nn);
  tg_count<<<(Ee + 255) / 256, 256, 0, stream>>>(ei, count, Ee);
  tg_scan<<<1, 256, 0, stream>>>(count, row_start, cursor, dinv, Nn, Ee);
  tg_fill<<<(Ee + 255) / 256, 256, 0, stream>>>(ei, cursor, col, Ee);
  tg_prep<<<1, 512, 0, stream>>>(czw, czb, lzw, lzb, chw, chb, lhw, lhb,
                                 attn, Wcomb, biasC, probs);

  const int rows = Nn * BB;                 // 80000
  const int ublocks = rows / 128;           // 625 (8 waves x 16 rows per block)
  for (int p = 0; p < PP; ++p) {
    tg_u<<<ublocks, 256, 0, stream>>>(x, Wcomb, U, Nn, p);
    tg_prop<<<Nn, 128, 0, stream>>>(U, row_start, col, dinv, biasC, probs, hacc, p);
  }
  tg_final<<<(Nn * PP + 255) / 256, 256, 0, stream>>>(hacc, outw, outb, out, Nn);
  (void)n_in; (void)out_size; (void)ws_size;
}